// GraphSAGEModel_82532091560100
// MI455X (gfx1250) — compile-verified
//
#include <hip/hip_runtime.h>
#include <hip/hip_bf16.h>

#define N_NODES   100000
#define N_FEAT    128
#define N_CLASSES 9
#define N_EDGES   1600000

typedef float v2f __attribute__((ext_vector_type(2)));
typedef float v8f __attribute__((ext_vector_type(8)));

// ---------------------------------------------------------------- zero ws
__global__ void sage_zero(float* __restrict__ p, int n) {
    int i = blockIdx.x * blockDim.x + threadIdx.x;
    if (i < n) p[i] = 0.0f;
}

// ---------------------------------------------------------------- scatter
// One wave32 per edge: lane owns 4 consecutive floats (16B) of the 128-float
// row. Gather x[src] is a coalesced 512B read; 32 fp32 atomics hit one 512B
// region of summed[dst] which lives in L2 (51.2MB < 192MB L2).
__global__ void sage_scatter(const float* __restrict__ x,
                             const long long* __restrict__ ei,
                             float* __restrict__ summed,
                             float* __restrict__ deg) {
    int tid = blockIdx.x * blockDim.x + threadIdx.x;
    int e = tid >> 5;
    int chunk = tid & 31;
    if (e >= N_EDGES) return;
    int src = (int)ei[e];
    int dst = (int)ei[N_EDGES + e];
    const float4 v = *(const float4*)(x + (long long)src * N_FEAT + chunk * 4);
    float* sp = summed + (long long)dst * N_FEAT + chunk * 4;
    (void)__hip_atomic_fetch_add(sp + 0, v.x, __ATOMIC_RELAXED, __HIP_MEMORY_SCOPE_AGENT);
    (void)__hip_atomic_fetch_add(sp + 1, v.y, __ATOMIC_RELAXED, __HIP_MEMORY_SCOPE_AGENT);
    (void)__hip_atomic_fetch_add(sp + 2, v.z, __ATOMIC_RELAXED, __HIP_MEMORY_SCOPE_AGENT);
    (void)__hip_atomic_fetch_add(sp + 3, v.w, __ATOMIC_RELAXED, __HIP_MEMORY_SCOPE_AGENT);
    if (chunk == 0)
        (void)__hip_atomic_fetch_add(deg + dst, 1.0f, __ATOMIC_RELAXED, __HIP_MEMORY_SCOPE_AGENT);
}

// ---------------------------------------------------------------- fused GEMM
// out = [aggr ; x] @ [W_l ; W_r] + b_l, then log_softmax over 9 classes.
// One wave = 16 nodes (M=16 tile), K=256 via 64x V_WMMA_F32_16X16X4_F32.
// B fragments (4x16, cols 9..15 zero) pre-staged in LDS in wmma lane layout.
__global__ __launch_bounds__(256)
void sage_gemm(const float* __restrict__ x,
               const float* __restrict__ Wl,
               const float* __restrict__ Wr,
               const float* __restrict__ bl,
               const float* __restrict__ summed,
               const float* __restrict__ deg,
               float* __restrict__ out) {
    __shared__ float ldsB[64 * 64];        // 64 k-steps x 32 lanes x 2 floats
    __shared__ float ldsT[8 * 16 * 17];    // per-wave 16x16 D tile, padded

    const int t = threadIdx.x;

    // Stage B fragments: ldsB[i*64 + lane*2 + part] = Wcat[4*(i%32)+koff+part][lane%16]
    for (int idx = t; idx < 64 * 64; idx += 256) {
        int i    = idx >> 6;
        int lane = (idx >> 1) & 31;
        int part = idx & 1;
        int koff = (lane >= 16) ? 2 : 0;
        int kl   = 4 * (i & 31) + koff + part;
        int n    = lane & 15;
        float v = 0.0f;
        if (n < N_CLASSES)
            v = (i < 32) ? Wl[kl * N_CLASSES + n] : Wr[kl * N_CLASSES + n];
        ldsB[idx] = v;
    }
    __syncthreads();

    const int wave = t >> 5;
    const int lane = t & 31;
    const int base = blockIdx.x * 128 + wave * 16;
    const int row  = base + (lane & 15);
    const int rowc = (row < N_NODES) ? row : (N_NODES - 1);  // clamp: EXEC must stay all-1s
    const int koff = (lane >= 16) ? 2 : 0;

    const float invd = 1.0f / fmaxf(deg[rowc], 1.0f);
    const float* __restrict__ arow = summed + (long long)rowc * N_FEAT;
    const float* __restrict__ xrow = x      + (long long)rowc * N_FEAT;

    v8f c = {};
    // K-half 1: aggr = summed * invdeg  vs  W_l
    for (int i = 0; i < 32; ++i) {
        v2f av = *(const v2f*)(arow + 4 * i + koff);
        v2f a; a.x = av.x * invd; a.y = av.y * invd;
        v2f b = *(const v2f*)(ldsB + i * 64 + lane * 2);
        c = __builtin_amdgcn_wmma_f32_16x16x4_f32(false, a, false, b,
                                                  (short)0, c, false, false);
    }
    // K-half 2: x vs W_r
    for (int i = 0; i < 32; ++i) {
        v2f a = *(const v2f*)(xrow + 4 * i + koff);
        v2f b = *(const v2f*)(ldsB + (i + 32) * 64 + lane * 2);
        c = __builtin_amdgcn_wmma_f32_16x16x4_f32(false, a, false, b,
                                                  (short)0, c, false, false);
    }

    // Spill D tile to LDS (row stride 17 avoids mod-64 bank aliasing)
    float* __restrict__ tp = ldsT + wave * (16 * 17);
    const int rlo = (lane < 16) ? 0 : 8;
    const int col = lane & 15;
#pragma unroll
    for (int v = 0; v < 8; ++v)
        tp[(rlo + v) * 17 + col] = c[v];
    __syncthreads();

    // 16 lanes per wave: bias + log_softmax over 9 classes, store
    if (lane < 16) {
        int r = base + lane;
        if (r < N_NODES) {
            float vals[N_CLASSES];
            float m = -INFINITY;
#pragma unroll
            for (int cc = 0; cc < N_CLASSES; ++cc) {
                vals[cc] = tp[lane * 17 + cc] + bl[cc];
                m = fmaxf(m, vals[cc]);
            }
            float s = 0.0f;
#pragma unroll
            for (int cc = 0; cc < N_CLASSES; ++cc)
                s += expf(vals[cc] - m);
            float lse = m + logf(s);
            float* op = out + (long long)r * N_CLASSES;
#pragma unroll
            for (int cc = 0; cc < N_CLASSES; ++cc)
                op[cc] = vals[cc] - lse;
        }
    }
}

// ---------------------------------------------------------------- launcher
extern "C" void kernel_launch(void* const* d_in, const int* in_sizes, int n_in,
                              void* d_out, int out_size, void* d_ws, size_t ws_size,
                              hipStream_t stream) {
    const float*     x  = (const float*)d_in[0];
    const long long* ei = (const long long*)d_in[1];   // int64 edge_index [2, E]
    const float*     Wl = (const float*)d_in[2];
    const float*     Wr = (const float*)d_in[3];
    const float*     bl = (const float*)d_in[4];
    float* out = (float*)d_out;

    // Workspace: summed [N_NODES*N_FEAT] | deg [N_NODES]  (~51.6 MB)
    float* summed = (float*)d_ws;
    float* deg    = summed + (size_t)N_NODES * N_FEAT;

    const int nzero = N_NODES * N_FEAT + N_NODES;
    sage_zero<<<(nzero + 255) / 256, 256, 0, stream>>>(summed, nzero);

    const int nthr = N_EDGES * 32;  // one wave per edge
    sage_scatter<<<(nthr + 255) / 256, 256, 0, stream>>>(x, ei, summed, deg);

    sage_gemm<<<(N_NODES + 127) / 128, 256, 0, stream>>>(x, Wl, Wr, bl,
                                                         summed, deg, out);
}